// ListenerModelBertAttCtxHist_86895778332805
// MI455X (gfx1250) — compile-verified
//
#include <hip/hip_runtime.h>
#include <hip/hip_bf16.h>

typedef __attribute__((ext_vector_type(2))) float v2f;
typedef __attribute__((ext_vector_type(8))) float v8f;

static __device__ __forceinline__ v8f wmma4(v2f a, v2f b, v8f c) {
    // D = A(16x4,f32) x B(4x16,f32) + C(16x16,f32)
    return __builtin_amdgcn_wmma_f32_16x16x4_f32(false, a, false, b, (short)0, c,
                                                 false, false);
}

// ---------------------------------------------------------------------------
// Weight repack: Wp[kp*N + n] = { W[(2kp)*N + n], W[(2kp+1)*N + n] }
// Turns the per-lane B-operand fetch into a single aligned b64 load.
// ---------------------------------------------------------------------------
__global__ __launch_bounds__(256) void pack_pairs(const float* __restrict__ W,
                                                  v2f* __restrict__ Wp,
                                                  int N, int total /* (K/2)*N */) {
    int idx = blockIdx.x * 256 + threadIdx.x;
    if (idx >= total) return;
    int kp = idx / N;
    int n = idx - kp * N;
    v2f p = {W[(size_t)(2 * kp) * N + n], W[(size_t)(2 * kp + 1) * N + n]};
    Wp[idx] = p;
}

// ---------------------------------------------------------------------------
// Generic fp32 WMMA GEMM: out[M,N] = act(A[M,K] @ W[K,N] + bias[N])
// W supplied pre-packed (pair-interleaved). One wave per 16(M) x 64(N) tile.
// ---------------------------------------------------------------------------
template <bool RELU>
__global__ __launch_bounds__(32) void gemm16(const float* __restrict__ A, int lda,
                                             const v2f* __restrict__ Wp,
                                             const float* __restrict__ bias,
                                             float* __restrict__ out, int N, int K) {
    const int lane = threadIdx.x;
    const int ml = lane & 15;            // M within tile (A), N within 16-tile (B/C/D)
    const int kh = (lane >> 4) << 1;     // K sub-offset: 0 or 2
    const int m0 = blockIdx.x << 4;
    const int n0 = blockIdx.y << 6;

    v8f acc[4] = {v8f{}, v8f{}, v8f{}, v8f{}};
    const float* Ar = A + (size_t)(m0 + ml) * lda;
    for (int k = 0; k < K; k += 4) {
        v2f a = *(const v2f*)(Ar + k + kh);
        const v2f* Wb = Wp + (size_t)((k + kh) >> 1) * N + n0 + ml;
        v2f b0 = Wb[0];
        v2f b1 = Wb[16];
        v2f b2 = Wb[32];
        v2f b3 = Wb[48];
        acc[0] = wmma4(a, b0, acc[0]);
        acc[1] = wmma4(a, b1, acc[1]);
        acc[2] = wmma4(a, b2, acc[2]);
        acc[3] = wmma4(a, b3, acc[3]);
    }
    // C/D layout: element j of acc[t] lives at row m0 + j + 8*(lane>>4), col n0+t*16+ml
    const int mBase = m0 + ((lane >> 4) << 3);
#pragma unroll
    for (int t = 0; t < 4; ++t) {
        int n = n0 + t * 16 + ml;
        float bs = bias[n];
#pragma unroll
        for (int j = 0; j < 8; ++j) {
            float v = acc[t][j] + bs;
            if (RELU) v = fmaxf(v, 0.0f);
            out[(size_t)(mBase + j) * N + n] = v;
        }
    }
}

// ---------------------------------------------------------------------------
// Fused chain: per 16-row tile of M = B*S (tiles never cross a batch):
//  L1: ir = relu(X @ W_e2h + b_e2h)          -> LDS (padded stride)
//  L2: mm = relu(ir @ W_mm[:512] + ctxW[b])  -> global mm_out
//  L3: score_m = sum_n tanh(mm @ W_a1 + b_a1)[m,n] * W_a2[n]  (+ b_a2)
// All weights pre-packed pair-interleaved.
// ---------------------------------------------------------------------------
#define LSTR 516  // 516 % 64 = 4 -> 16 A-rows land on distinct bank groups

__global__ __launch_bounds__(32) void fused_chain(
    const float* __restrict__ X,                                     // [B*S,768]
    const v2f* __restrict__ W_e2h_p, const float* __restrict__ b_e2h,
    const v2f* __restrict__ W_mm1_p,                                 // rows 0..511
    const float* __restrict__ ctxW,                                  // [B,512]
    const v2f* __restrict__ W_a1_p, const float* __restrict__ b_a1,
    const float* __restrict__ W_a2, const float* __restrict__ b_a2,
    float* __restrict__ mm_out,                                      // [B*S,512]
    float* __restrict__ scores) {                                    // [B*S]
    __shared__ float ldsIR[16 * LSTR];
    const int lane = threadIdx.x;
    const int ml = lane & 15;
    const int kh = (lane >> 4) << 1;
    const int mhalf = (lane >> 4) << 3;  // 0 or 8
    const int row0 = blockIdx.x << 4;
    const int batch = row0 >> 9;         // 32 tiles per batch (S=512)

    // ---- Layer 1: e2h + relu -> LDS ----
    const float* Ar = X + (size_t)(row0 + ml) * 768;
    for (int ng = 0; ng < 32; ng += 4) {
        v8f acc[4] = {v8f{}, v8f{}, v8f{}, v8f{}};
        for (int k = 0; k < 768; k += 4) {
            v2f a = *(const v2f*)(Ar + k + kh);
            const v2f* Wb = W_e2h_p + (size_t)((k + kh) >> 1) * 512 + (ng << 4) + ml;
            v2f b0 = Wb[0];
            v2f b1 = Wb[16];
            v2f b2 = Wb[32];
            v2f b3 = Wb[48];
            acc[0] = wmma4(a, b0, acc[0]);
            acc[1] = wmma4(a, b1, acc[1]);
            acc[2] = wmma4(a, b2, acc[2]);
            acc[3] = wmma4(a, b3, acc[3]);
        }
#pragma unroll
        for (int t = 0; t < 4; ++t) {
            int n = (ng + t) * 16 + ml;
            float bs = b_e2h[n];
#pragma unroll
            for (int j = 0; j < 8; ++j)
                ldsIR[(mhalf + j) * LSTR + n] = fmaxf(acc[t][j] + bs, 0.0f);
        }
    }
    __syncthreads();

    // ---- Layer 2: mm = relu(ir @ W_mm1 + ctxW[b]) -> global ----
    const float* cxb = ctxW + (size_t)batch * 512;
    for (int ng = 0; ng < 32; ng += 4) {
        v8f acc[4] = {v8f{}, v8f{}, v8f{}, v8f{}};
        for (int k = 0; k < 512; k += 4) {
            v2f a = *(const v2f*)&ldsIR[ml * LSTR + k + kh];
            const v2f* Wb = W_mm1_p + (size_t)((k + kh) >> 1) * 512 + (ng << 4) + ml;
            v2f b0 = Wb[0];
            v2f b1 = Wb[16];
            v2f b2 = Wb[32];
            v2f b3 = Wb[48];
            acc[0] = wmma4(a, b0, acc[0]);
            acc[1] = wmma4(a, b1, acc[1]);
            acc[2] = wmma4(a, b2, acc[2]);
            acc[3] = wmma4(a, b3, acc[3]);
        }
#pragma unroll
        for (int t = 0; t < 4; ++t) {
            int n = (ng + t) * 16 + ml;
            float ex = cxb[n];
#pragma unroll
            for (int j = 0; j < 8; ++j) {
                float v = fmaxf(acc[t][j] + ex, 0.0f);
                mm_out[(size_t)(row0 + mhalf + j) * 512 + n] = v;
            }
        }
    }

    // ---- Layer 3 + score (same-wave store->load to mm_out is kept in order) ----
    const float* Mr = mm_out + (size_t)(row0 + ml) * 512;
    float sacc[8] = {0.f, 0.f, 0.f, 0.f, 0.f, 0.f, 0.f, 0.f};
    for (int ng = 0; ng < 32; ng += 4) {
        v8f acc[4] = {v8f{}, v8f{}, v8f{}, v8f{}};
        for (int k = 0; k < 512; k += 4) {
            v2f a = *(const v2f*)(Mr + k + kh);
            const v2f* Wb = W_a1_p + (size_t)((k + kh) >> 1) * 512 + (ng << 4) + ml;
            v2f b0 = Wb[0];
            v2f b1 = Wb[16];
            v2f b2 = Wb[32];
            v2f b3 = Wb[48];
            acc[0] = wmma4(a, b0, acc[0]);
            acc[1] = wmma4(a, b1, acc[1]);
            acc[2] = wmma4(a, b2, acc[2]);
            acc[3] = wmma4(a, b3, acc[3]);
        }
#pragma unroll
        for (int t = 0; t < 4; ++t) {
            int n = (ng + t) * 16 + ml;
            float ba = b_a1[n];
            float wa2 = W_a2[n];
#pragma unroll
            for (int j = 0; j < 8; ++j)
                sacc[j] += tanhf(acc[t][j] + ba) * wa2;
        }
    }
    float ba2 = b_a2[0];
#pragma unroll
    for (int j = 0; j < 8; ++j) {
        float v = sacc[j];
        v += __shfl_xor(v, 1, 32);
        v += __shfl_xor(v, 2, 32);
        v += __shfl_xor(v, 4, 32);
        v += __shfl_xor(v, 8, 32);  // reduce within each 16-lane half
        if (ml == j) scores[row0 + mhalf + j] = v + ba2;
    }
}

// ---------------------------------------------------------------------------
// Per-batch masked softmax over S + attended[b,h] = sum_s mm[b,s,h]*w[s]
// ---------------------------------------------------------------------------
__global__ __launch_bounds__(256) void attend_kernel(
    const float* __restrict__ scores, const unsigned char* __restrict__ masks,
    const float* __restrict__ mm, float* __restrict__ attended) {
    __shared__ float red[256];
    __shared__ float w[512];
    const int b = blockIdx.x, t = threadIdx.x;
    const float* sc = scores + (size_t)b * 512;
    const unsigned char* mk = masks + (size_t)b * 512;

    float v0 = mk[t] ? -__builtin_inff() : sc[t];
    float v1 = mk[t + 256] ? -__builtin_inff() : sc[t + 256];
    red[t] = fmaxf(v0, v1);
    __syncthreads();
    for (int s = 128; s > 0; s >>= 1) {
        if (t < s) red[t] = fmaxf(red[t], red[t + s]);
        __syncthreads();
    }
    float mx = red[0];
    __syncthreads();
    float e0 = expf(v0 - mx), e1 = expf(v1 - mx);
    w[t] = e0;
    w[t + 256] = e1;
    red[t] = e0 + e1;
    __syncthreads();
    for (int s = 128; s > 0; s >>= 1) {
        if (t < s) red[t] += red[t + s];
        __syncthreads();
    }
    float inv = 1.0f / red[0];
    w[t] *= inv;
    w[t + 256] *= inv;
    __syncthreads();

    const int h = t << 1;
    const float* mb = mm + (size_t)b * 512 * 512 + h;
    float a0 = 0.f, a1 = 0.f;
    for (int s = 0; s < 512; ++s) {
        float wt = w[s];
        float2 mv = *(const float2*)(mb + (size_t)s * 512);
        a0 = fmaf(wt, mv.x, a0);
        a1 = fmaf(wt, mv.y, a1);
    }
    attended[(size_t)b * 512 + h] = a0;
    attended[(size_t)b * 512 + h + 1] = a1;
}

// ---------------------------------------------------------------------------
// History masked average: havg[bk,:] = sum_{hl<cnt} prev_hist[bk,hl,:] / max(cnt,1)
// ---------------------------------------------------------------------------
__global__ __launch_bounds__(256) void hist_avg_kernel(
    const float* __restrict__ prev_hist, const int* __restrict__ counts,
    float* __restrict__ havg) {
    const int bk = blockIdx.x;
    const int cnt = counts[bk];
    const float inv = 1.0f / (float)(cnt < 1 ? 1 : cnt);
    const float* base = prev_hist + (size_t)bk * 20 * 768;
    for (int e = threadIdx.x; e < 768; e += 256) {
        float s = 0.f;
        for (int h = 0; h < cnt; ++h) s += base[(size_t)h * 768 + e];
        havg[(size_t)bk * 768 + e] = s * inv;
    }
}

// ---------------------------------------------------------------------------
// sep = relu(sep_raw + (cnt>0 ? hproj : 0)); L2-normalize over H; dot attended
// ---------------------------------------------------------------------------
__global__ __launch_bounds__(256) void finalize_kernel(
    const float* __restrict__ sep_raw, const float* __restrict__ hproj,
    const int* __restrict__ counts, const float* __restrict__ attended,
    float* __restrict__ out) {
    __shared__ float red[256];
    const int bk = blockIdx.x, t = threadIdx.x;
    const int b = bk / 6;
    const int cnt = counts[bk];
    float v0 = sep_raw[(size_t)bk * 512 + t];
    float v1 = sep_raw[(size_t)bk * 512 + 256 + t];
    if (cnt > 0) {
        v0 += hproj[(size_t)bk * 512 + t];
        v1 += hproj[(size_t)bk * 512 + 256 + t];
    }
    v0 = fmaxf(v0, 0.f);
    v1 = fmaxf(v1, 0.f);
    red[t] = v0 * v0 + v1 * v1;
    __syncthreads();
    for (int s = 128; s > 0; s >>= 1) {
        if (t < s) red[t] += red[t + s];
        __syncthreads();
    }
    float norm = fmaxf(sqrtf(red[0]), 1e-12f);
    __syncthreads();
    red[t] = v0 * attended[(size_t)b * 512 + t] +
             v1 * attended[(size_t)b * 512 + 256 + t];
    __syncthreads();
    for (int s = 128; s > 0; s >>= 1) {
        if (t < s) red[t] += red[t + s];
        __syncthreads();
    }
    if (t == 0) out[bk] = red[0] / norm;
}

// ---------------------------------------------------------------------------
extern "C" void kernel_launch(void* const* d_in, const int* in_sizes, int n_in,
                              void* d_out, int out_size, void* d_ws, size_t ws_size,
                              hipStream_t stream) {
    const float* reps       = (const float*)d_in[0];   // [64,512,768]
    // d_in[1] = lengths (unused by reference)
    const float* sep_img    = (const float*)d_in[2];   // [64,6,2048]
    const float* vctx       = (const float*)d_in[3];   // [64,12288]
    const float* prev_hist  = (const float*)d_in[4];   // [64,6,20,768]
    const int*   hist_cnts  = (const int*)d_in[5];     // [64,6]
    const unsigned char* masks = (const unsigned char*)d_in[6];  // [64,512,1] bool
    const float* W_sep = (const float*)d_in[7],  *b_sep = (const float*)d_in[8];
    const float* W_e2h = (const float*)d_in[9],  *b_e2h = (const float*)d_in[10];
    const float* W_hist= (const float*)d_in[11], *b_hist= (const float*)d_in[12];
    const float* W_ctx = (const float*)d_in[13], *b_ctx = (const float*)d_in[14];
    const float* W_mm  = (const float*)d_in[15], *b_mm  = (const float*)d_in[16];
    const float* W_a1  = (const float*)d_in[17], *b_a1  = (const float*)d_in[18];
    const float* W_a2  = (const float*)d_in[19], *b_a2  = (const float*)d_in[20];

    // workspace layout (floats)
    float* ws       = (float*)d_ws;
    float* ctx      = ws;                   // 64*512
    float* ctxW     = ctx      + 32768;     // 64*512
    float* havg     = ctxW     + 32768;     // 384*768
    float* hproj    = havg     + 294912;    // 384*512
    float* sep_raw  = hproj    + 196608;    // 384*512
    float* attended = sep_raw  + 196608;    // 64*512
    float* scoresb  = attended + 32768;     // 64*512
    float* mm_buf   = scoresb  + 32768;     // 32768*512 (64 MB)
    float* packs    = mm_buf   + 16777216;
    v2f* W_ctx_p  = (v2f*)packs;                 // 12288*512 floats
    v2f* W_e2h_p  = (v2f*)(packs + 6291456);     //   768*512
    v2f* W_mm_p   = (v2f*)(packs + 6684672);     //  1024*512
    v2f* W_a1_p   = (v2f*)(packs + 7208960);     //   512*512
    v2f* W_hist_p = (v2f*)(packs + 7471104);     //   768*512
    v2f* W_sep_p  = (v2f*)(packs + 7864320);     //  2048*512

    dim3 wave(32), blk(256);

    // 0) pack all weights into pair-interleaved form (one b64 per B operand)
    auto packN = [&](const float* W, v2f* Wp, int K) {
        int total = (K / 2) * 512;
        pack_pairs<<<(total + 255) / 256, blk, 0, stream>>>(W, Wp, 512, total);
    };
    packN(W_ctx, W_ctx_p, 12288);
    packN(W_e2h, W_e2h_p, 768);
    packN(W_mm, W_mm_p, 1024);
    packN(W_a1, W_a1_p, 512);
    packN(W_hist, W_hist_p, 768);
    packN(W_sep, W_sep_p, 2048);

    // 1) ctx = relu(vctx @ W_ctx + b_ctx)            M=64 N=512 K=12288
    gemm16<true><<<dim3(4, 8), wave, 0, stream>>>(vctx, 12288, W_ctx_p, b_ctx, ctx, 512, 12288);
    // 2) ctxW = ctx @ W_mm[512:,:] + b_mm            M=64 N=512 K=512 (packed rows 512..1023)
    gemm16<false><<<dim3(4, 8), wave, 0, stream>>>(ctx, 512, W_mm_p + 256 * 512, b_mm, ctxW, 512, 512);
    // 3) history average                              384 rows
    hist_avg_kernel<<<384, blk, 0, stream>>>(prev_hist, hist_cnts, havg);
    // 4) hproj = relu(havg @ W_hist + b_hist)        M=384 N=512 K=768
    gemm16<true><<<dim3(24, 8), wave, 0, stream>>>(havg, 768, W_hist_p, b_hist, hproj, 512, 768);
    // 5) sep_raw = sep_img @ W_sep + b_sep           M=384 N=512 K=2048
    gemm16<false><<<dim3(24, 8), wave, 0, stream>>>(sep_img, 2048, W_sep_p, b_sep, sep_raw, 512, 2048);
    // 6) fused e2h -> mm -> tanh-attention scores    M=32768 (2048 tiles)
    fused_chain<<<2048, wave, 0, stream>>>(reps, W_e2h_p, b_e2h, W_mm_p, ctxW,
                                           W_a1_p, b_a1, W_a2, b_a2, mm_buf, scoresb);
    // 7) masked softmax + attended
    attend_kernel<<<64, blk, 0, stream>>>(scoresb, masks, mm_buf, attended);
    // 8) combine, normalize, dot -> out [64,6,1]
    finalize_kernel<<<384, blk, 0, stream>>>(sep_raw, hproj, hist_cnts, attended,
                                             (float*)d_out);
}